// GraphSAGE_76673756168224
// MI455X (gfx1250) — compile-verified
//
#include <hip/hip_runtime.h>

typedef __attribute__((ext_vector_type(2))) float v2f;
typedef __attribute__((ext_vector_type(8))) float v8f;

#define IN_C 128
#define HID  128
#define OUT_C 64

// ---------------------------------------------------------------- utilities
__global__ void zero_kernel(float* __restrict__ p, size_t n) {
    size_t i = (size_t)blockIdx.x * blockDim.x + threadIdx.x;
    size_t stride = (size_t)gridDim.x * blockDim.x;
    for (; i < n; i += stride) p[i] = 0.0f;
}

__global__ void deg_kernel(const int* __restrict__ dst, float* __restrict__ deg, int E) {
    int e = blockIdx.x * blockDim.x + threadIdx.x;
    if (e < E) atomicAdd(&deg[dst[e]], 1.0f);
}

__global__ void dinv_kernel(float* __restrict__ deg, int N) {
    int i = blockIdx.x * blockDim.x + threadIdx.x;
    if (i < N) deg[i] = 1.0f / fmaxf(deg[i], 1.0f);
}

// One wave (32 lanes) per edge; each lane scatters 4 contiguous floats.
__global__ void aggr_kernel(const float* __restrict__ x,
                            const int* __restrict__ src,
                            const int* __restrict__ dst,
                            float* __restrict__ aggr, int E) {
    int t = blockIdx.x * blockDim.x + threadIdx.x;
    int e = t >> 5;
    if (e >= E) return;
    int lane = t & 31;
    int s = src[e];
    int d = dst[e];
    const float4 v = *(const float4*)(x + (size_t)s * IN_C + lane * 4);
    float* o = aggr + (size_t)d * IN_C + lane * 4;
    atomicAdd(o + 0, v.x);
    atomicAdd(o + 1, v.y);
    atomicAdd(o + 2, v.z);
    atomicAdd(o + 3, v.w);
}

// -------------------------------------------------- fused SAGE layer (WMMA)
// out[16 rows x 128 cols] = relu( (aggr*dinv) @ Wl + x @ Wr + bias )
// One wave per 16-row tile; f32 WMMA 16x16x4, K = 128.
__global__ __launch_bounds__(256)
void sage_wmma_kernel(const float* __restrict__ aggr,
                      const float* __restrict__ dinv,
                      const float* __restrict__ x,
                      const float* __restrict__ Wl,
                      const float* __restrict__ bias,
                      const float* __restrict__ Wr,
                      float* __restrict__ out,
                      int N, int nRowTiles) {
    int wave = threadIdx.x >> 5;
    int lane = threadIdx.x & 31;
    int tile = blockIdx.x * (blockDim.x >> 5) + wave;
    if (tile >= nRowTiles) return;                 // wave-uniform: EXEC stays all-1s

    int row0  = tile * 16;
    int mrow  = lane & 15;                         // A-matrix row this lane owns
    int khalf = (lane >> 4) * 2;                   // lanes 0-15: K offs 0,1; 16-31: 2,3
    int arow  = min(row0 + mrow, N - 1);
    float dv  = dinv[arow];
    const float* aRow = aggr + (size_t)arow * HID;
    const float* xRow = x    + (size_t)arow * HID;
    const bool fullTile = (row0 + 16 <= N);        // wave-uniform

    for (int nt = 0; nt < HID / 16; ++nt) {
        int n0  = nt * 16;
        int col = n0 + mrow;                       // C/D: N = lane&15
        float bv = bias[col];
        v8f acc = {bv, bv, bv, bv, bv, bv, bv, bv};

        for (int k0 = 0; k0 < HID; k0 += 4) {
            int k = k0 + khalf;
            v2f a1; a1.x = aRow[k] * dv;  a1.y = aRow[k + 1] * dv;
            v2f b1; b1.x = Wl[(size_t)k * HID + col];
                    b1.y = Wl[(size_t)(k + 1) * HID + col];
            acc = __builtin_amdgcn_wmma_f32_16x16x4_f32(
                      false, a1, false, b1, (short)0, acc, false, false);
            v2f a2; a2.x = xRow[k];       a2.y = xRow[k + 1];
            v2f b2; b2.x = Wr[(size_t)k * HID + col];
                    b2.y = Wr[(size_t)(k + 1) * HID + col];
            acc = __builtin_amdgcn_wmma_f32_16x16x4_f32(
                      false, a2, false, b2, (short)0, acc, false, false);
        }

        int rbase = row0 + (lane >> 4) * 8;        // C/D: VGPR r -> M = r + 8*(lane>>4)
        if (fullTile) {
            // branch-free coalesced stores (always taken: N % 16 == 0 here)
#pragma unroll
            for (int r = 0; r < 8; ++r) {
                float v = acc[r];
                out[(size_t)(rbase + r) * HID + col] = v > 0.0f ? v : 0.0f;
            }
        } else {
#pragma unroll
            for (int r = 0; r < 8; ++r) {
                int row = rbase + r;
                if (row < N) {
                    float v = acc[r];
                    out[(size_t)row * HID + col] = v > 0.0f ? v : 0.0f;
                }
            }
        }
    }
}

// ------------------------------------------------------- classifier (WMMA)
// logits[16 x 64] = h @ Wc + bc   (K = 128, no relu)
__global__ __launch_bounds__(256)
void cls_wmma_kernel(const float* __restrict__ h,
                     const float* __restrict__ Wc,
                     const float* __restrict__ bc,
                     float* __restrict__ out,
                     int N, int nRowTiles) {
    int wave = threadIdx.x >> 5;
    int lane = threadIdx.x & 31;
    int tile = blockIdx.x * (blockDim.x >> 5) + wave;
    if (tile >= nRowTiles) return;

    int row0  = tile * 16;
    int mrow  = lane & 15;
    int khalf = (lane >> 4) * 2;
    int arow  = min(row0 + mrow, N - 1);
    const float* hRow = h + (size_t)arow * HID;
    const bool fullTile = (row0 + 16 <= N);

    for (int nt = 0; nt < OUT_C / 16; ++nt) {
        int n0  = nt * 16;
        int col = n0 + mrow;
        float bv = bc[col];
        v8f acc = {bv, bv, bv, bv, bv, bv, bv, bv};

        for (int k0 = 0; k0 < HID; k0 += 4) {
            int k = k0 + khalf;
            v2f a; a.x = hRow[k];  a.y = hRow[k + 1];
            v2f b; b.x = Wc[(size_t)k * OUT_C + col];
                   b.y = Wc[(size_t)(k + 1) * OUT_C + col];
            acc = __builtin_amdgcn_wmma_f32_16x16x4_f32(
                      false, a, false, b, (short)0, acc, false, false);
        }

        int rbase = row0 + (lane >> 4) * 8;
        if (fullTile) {
#pragma unroll
            for (int r = 0; r < 8; ++r)
                out[(size_t)(rbase + r) * OUT_C + col] = acc[r];
        } else {
#pragma unroll
            for (int r = 0; r < 8; ++r) {
                int row = rbase + r;
                if (row < N) out[(size_t)row * OUT_C + col] = acc[r];
            }
        }
    }
}

// ------------------------------------------------------------------ driver
extern "C" void kernel_launch(void* const* d_in, const int* in_sizes, int n_in,
                              void* d_out, int out_size, void* d_ws, size_t ws_size,
                              hipStream_t stream) {
    const float* x_doc = (const float*)d_in[0];
    // d_in[1] x_token: unused by reference
    const int*   eidx  = (const int*)d_in[2];
    // d_in[3] edge_weight: unused by reference
    const float* W1l = (const float*)d_in[4];
    const float* b1  = (const float*)d_in[5];
    const float* W1r = (const float*)d_in[6];
    const float* W2l = (const float*)d_in[7];
    const float* b2  = (const float*)d_in[8];
    const float* W2r = (const float*)d_in[9];
    const float* Wc  = (const float*)d_in[10];
    const float* bc  = (const float*)d_in[11];

    const int N = in_sizes[0] / IN_C;
    const int E = in_sizes[2] / 2;
    const int* src = eidx;
    const int* dst = eidx + E;

    // workspace layout: deg_inv | aggr(N*128) | h1(N*128)
    float* deg  = (float*)d_ws;
    size_t off  = (((size_t)N * sizeof(float)) + 255) & ~(size_t)255;
    float* aggr = (float*)((char*)d_ws + off);
    float* h1   = aggr + (size_t)N * HID;

    float* logits = (float*)d_out;               // N x 64
    float* h2     = logits + (size_t)N * OUT_C;  // N x 128 (second tuple output)

    const int nRowTiles = (N + 15) / 16;
    dim3 blk(256);
    dim3 gemmGrid((nRowTiles + 7) / 8);
    dim3 edgeGrid((E + 255) / 256);
    dim3 aggrGrid(((size_t)E * 32 + 255) / 256);
    dim3 nodeGrid((N + 255) / 256);

    // degrees
    zero_kernel<<<2048, blk, 0, stream>>>(deg, (size_t)N);
    deg_kernel<<<edgeGrid, blk, 0, stream>>>(dst, deg, E);
    dinv_kernel<<<nodeGrid, blk, 0, stream>>>(deg, N);

    // layer 1
    zero_kernel<<<4096, blk, 0, stream>>>(aggr, (size_t)N * HID);
    aggr_kernel<<<aggrGrid, blk, 0, stream>>>(x_doc, src, dst, aggr, E);
    sage_wmma_kernel<<<gemmGrid, blk, 0, stream>>>(aggr, deg, x_doc, W1l, b1, W1r,
                                                   h1, N, nRowTiles);

    // layer 2 (output h straight into d_out slot)
    zero_kernel<<<4096, blk, 0, stream>>>(aggr, (size_t)N * HID);
    aggr_kernel<<<aggrGrid, blk, 0, stream>>>(h1, src, dst, aggr, E);
    sage_wmma_kernel<<<gemmGrid, blk, 0, stream>>>(aggr, deg, h1, W2l, b2, W2r,
                                                   h2, N, nRowTiles);

    // classifier
    cls_wmma_kernel<<<gemmGrid, blk, 0, stream>>>(h2, Wc, bc, logits, N, nRowTiles);
}